// CouplingGCN_16329465660189
// MI455X (gfx1250) — compile-verified
//
#include <hip/hip_runtime.h>

#define N_NODES 50000
#define N_EDGES 600000
#define N_PAIRS 200000
#define F_ATOM 64
#define F_PAIR 32
#define HID 128
#define N_LAYERS 3
#define EPS 1e-5f

typedef float v2f __attribute__((ext_vector_type(2)));
typedef float v8f __attribute__((ext_vector_type(8)));

// ---------------------------------------------------------------------------
// degree / normalization
// ---------------------------------------------------------------------------
__global__ void init_deg_kernel(float* deg, int n) {
    int i = blockIdx.x * blockDim.x + threadIdx.x;
    if (i < n) deg[i] = 1.0f;  // self-loop
}

__global__ void count_deg_kernel(const int* __restrict__ row, float* deg, int e) {
    int i = blockIdx.x * blockDim.x + threadIdx.x;
    if (i < e) atomicAdd(&deg[row[i]], 1.0f);
}

__global__ void dinv_kernel(const float* __restrict__ deg, float* dinv, int n) {
    int i = blockIdx.x * blockDim.x + threadIdx.x;
    if (i < n) dinv[i] = rsqrtf(deg[i]);  // deg >= 1 always (self-loop)
}

// single-block exclusive scan of integer degrees -> rowstart[0..n]
__global__ void scan_kernel(const float* __restrict__ deg, int* rowstart, int n) {
    __shared__ int buf[1024];
    __shared__ int running;
    int tid = threadIdx.x;
    if (tid == 0) running = 0;
    __syncthreads();
    for (int base = 0; base < n; base += 1024) {
        int i = base + tid;
        int c = (i < n) ? (int)(deg[i] + 0.5f) : 0;
        buf[tid] = c;
        __syncthreads();
        for (int off = 1; off < 1024; off <<= 1) {
            int v = (tid >= off) ? buf[tid - off] : 0;
            __syncthreads();
            buf[tid] += v;
            __syncthreads();
        }
        if (i < n) rowstart[i] = running + buf[tid] - c;  // exclusive
        __syncthreads();
        if (tid == 0) running += buf[1023];
        __syncthreads();
    }
    if (tid == 0) rowstart[n] = running;
}

// place self-loop first, init cursor
__global__ void csr_selfloop_kernel(const int* __restrict__ rowstart, int* cursor,
                                    int* col_sorted, int n) {
    int i = blockIdx.x * blockDim.x + threadIdx.x;
    if (i < n) {
        col_sorted[rowstart[i]] = i;
        cursor[i] = 1;
    }
}

__global__ void csr_fill_kernel(const int* __restrict__ row, const int* __restrict__ col,
                                const int* __restrict__ rowstart, int* cursor,
                                int* col_sorted, int e) {
    int i = blockIdx.x * blockDim.x + threadIdx.x;
    if (i < e) {
        int r = row[i];
        int p = atomicAdd(&cursor[r], 1);
        col_sorted[rowstart[r] + p] = col[i];
    }
}

// ---------------------------------------------------------------------------
// fp32 WMMA GEMM: out[M x 128] = A[M x K] @ W[K x 128] (+bias)
// one block = 256 threads = 8 waves; block handles one 16-row M tile,
// each wave one 16-col N tile. A tile staged in LDS (padded stride).
// ---------------------------------------------------------------------------
__global__ void gemm_xw_kernel(const float* __restrict__ A, const float* __restrict__ W,
                               const float* __restrict__ bias, float* __restrict__ out,
                               int K) {
    __shared__ float As[16 * 132];  // K<=128, padded stride K+4
    const int ldk = K + 4;
    const int mt = blockIdx.x;
    const int tid = threadIdx.x;

    for (int i = tid; i < 16 * K; i += 256) {
        int r = i / K, c = i - r * K;
        As[r * ldk + c] = A[(size_t)(mt * 16 + r) * K + c];
    }
    __syncthreads();

    const int wave = tid >> 5;
    const int lane = tid & 31;
    const int half = lane >> 4;   // 0: K pair (k,k+1); 1: (k+2,k+3)
    const int l = lane & 15;      // row (A) / col (B)
    const int n0 = wave * 16;

    v8f acc = {};
    for (int k = 0; k < K; k += 4) {
        v2f a, b;
        a[0] = As[l * ldk + k + 2 * half + 0];
        a[1] = As[l * ldk + k + 2 * half + 1];
        b[0] = W[(k + 2 * half + 0) * HID + n0 + l];
        b[1] = W[(k + 2 * half + 1) * HID + n0 + l];
        acc = __builtin_amdgcn_wmma_f32_16x16x4_f32(false, a, false, b,
                                                    (short)0, acc, false, false);
    }

    const int colg = n0 + l;
    const float bb = bias ? bias[colg] : 0.0f;
#pragma unroll
    for (int v = 0; v < 8; ++v) {
        int r = mt * 16 + v + 8 * half;
        out[(size_t)r * HID + colg] = acc[v] + bb;
    }
}

// ---------------------------------------------------------------------------
// GCN aggregation fused with bias + batchnorm + relu. One wave per node,
// float4 per lane covers HID=128. Gathers of h[j] stay in L2 (25.6MB).
// ---------------------------------------------------------------------------
__global__ void aggregate_bn_relu_kernel(const float* __restrict__ h,
                                         const int* __restrict__ rowstart,
                                         const int* __restrict__ cols,
                                         const float* __restrict__ dinv,
                                         const float* __restrict__ bias,
                                         const float* __restrict__ gamma,
                                         const float* __restrict__ beta,
                                         const float* __restrict__ mean,
                                         const float* __restrict__ var,
                                         float* __restrict__ xout, int n) {
    int node = (blockIdx.x * blockDim.x + threadIdx.x) >> 5;
    int lane = threadIdx.x & 31;
    if (node >= n) return;

    int s = rowstart[node], e = rowstart[node + 1];
    float di = dinv[node];
    float4 acc = make_float4(0.f, 0.f, 0.f, 0.f);
    for (int idx = s; idx < e; ++idx) {
        int j = cols[idx];
        float w = di * dinv[j];
        float4 hv = ((const float4*)(h + (size_t)j * HID))[lane];
        acc.x += hv.x * w; acc.y += hv.y * w;
        acc.z += hv.z * w; acc.w += hv.w * w;
    }
    int f = lane * 4;
    float* o = xout + (size_t)node * HID + f;
#pragma unroll
    for (int c = 0; c < 4; ++c) {
        float v = ((&acc.x)[c] + bias[f + c] - mean[f + c]) *
                  rsqrtf(var[f + c] + EPS) * gamma[f + c] + beta[f + c];
        o[c] = fmaxf(v, 0.0f);
    }
}

// ---------------------------------------------------------------------------
// fused pair MLP: gather [x[p0] | x[p1] | pf] (16x288) -> 128 relu -> 64 relu -> 1
// one block = 16 pairs, 8 waves, WMMA f32 16x16x4 throughout.
// ---------------------------------------------------------------------------
__global__ void pair_mlp_kernel(const float* __restrict__ x, const int* __restrict__ pairs,
                                const float* __restrict__ pf,
                                const float* __restrict__ W1, const float* __restrict__ b1,
                                const float* __restrict__ W2, const float* __restrict__ b2,
                                const float* __restrict__ W3, const float* __restrict__ b3,
                                float* __restrict__ out) {
    __shared__ float Z[16][292];   // 288 + pad
    __shared__ float H1[16][132];  // 128 + pad
    __shared__ float H2[16][68];   // 64 + pad

    const int pt = blockIdx.x;   // pair tile of 16
    const int tid = threadIdx.x;

    for (int i = tid; i < 16 * 288; i += 256) {
        int r = i / 288, c = i - r * 288;
        int p = pt * 16 + r;
        float v;
        if (c < HID)            v = x[(size_t)pairs[p * 2 + 0] * HID + c];
        else if (c < 2 * HID)   v = x[(size_t)pairs[p * 2 + 1] * HID + (c - HID)];
        else                    v = pf[(size_t)p * F_PAIR + (c - 2 * HID)];
        Z[r][c] = v;
    }
    __syncthreads();

    const int wave = tid >> 5;
    const int lane = tid & 31;
    const int half = lane >> 4;
    const int l = lane & 15;

    // ---- layer 1: 16x288 @ 288x128, relu ----
    {
        int n0 = wave * 16;
        v8f acc = {};
        for (int k = 0; k < 2 * HID + F_PAIR; k += 4) {
            v2f a, b;
            a[0] = Z[l][k + 2 * half + 0];
            a[1] = Z[l][k + 2 * half + 1];
            b[0] = W1[(k + 2 * half + 0) * HID + n0 + l];
            b[1] = W1[(k + 2 * half + 1) * HID + n0 + l];
            acc = __builtin_amdgcn_wmma_f32_16x16x4_f32(false, a, false, b,
                                                        (short)0, acc, false, false);
        }
#pragma unroll
        for (int v = 0; v < 8; ++v)
            H1[v + 8 * half][n0 + l] = fmaxf(acc[v] + b1[n0 + l], 0.0f);
    }
    __syncthreads();

    // ---- layer 2: 16x128 @ 128x64, relu (all 8 waves execute so EXEC stays
    // all-ones for WMMA; waves 4-7 duplicate 0-3, writing identical values) ----
    {
        int n0 = (wave & 3) * 16;
        v8f acc = {};
        for (int k = 0; k < HID; k += 4) {
            v2f a, b;
            a[0] = H1[l][k + 2 * half + 0];
            a[1] = H1[l][k + 2 * half + 1];
            b[0] = W2[(k + 2 * half + 0) * 64 + n0 + l];
            b[1] = W2[(k + 2 * half + 1) * 64 + n0 + l];
            acc = __builtin_amdgcn_wmma_f32_16x16x4_f32(false, a, false, b,
                                                        (short)0, acc, false, false);
        }
#pragma unroll
        for (int v = 0; v < 8; ++v)
            H2[v + 8 * half][n0 + l] = fmaxf(acc[v] + b2[n0 + l], 0.0f);
    }
    __syncthreads();

    // ---- layer 3: 16x64 @ 64x1 ----
    if (tid < 16) {
        float s = b3[0];
#pragma unroll
        for (int k = 0; k < 64; ++k) s += H2[tid][k] * W3[k];
        out[pt * 16 + tid] = s;
    }
}

// ---------------------------------------------------------------------------
// host-side orchestration
// ---------------------------------------------------------------------------
static inline size_t align256(size_t x) { return (x + 255) & ~(size_t)255; }

extern "C" void kernel_launch(void* const* d_in, const int* in_sizes, int n_in,
                              void* d_out, int out_size, void* d_ws, size_t ws_size,
                              hipStream_t stream) {
    const float* atom   = (const float*)d_in[0];
    const int*   eidx   = (const int*)d_in[1];   // (2, E): row then col
    const int*   pairs  = (const int*)d_in[2];   // (P, 2)
    const float* pfeat  = (const float*)d_in[3];
    const float* emW    = (const float*)d_in[4];
    const float* emB    = (const float*)d_in[5];
    const float* convW  = (const float*)d_in[6]; // (3,128,128)
    const float* convB  = (const float*)d_in[7];
    const float* gamma  = (const float*)d_in[8];
    const float* beta   = (const float*)d_in[9];
    const float* mean   = (const float*)d_in[10];
    const float* var    = (const float*)d_in[11];
    const float* W1     = (const float*)d_in[12];
    const float* b1     = (const float*)d_in[13];
    const float* W2     = (const float*)d_in[14];
    const float* b2     = (const float*)d_in[15];
    const float* W3     = (const float*)d_in[16];
    const float* b3     = (const float*)d_in[17];
    float* out = (float*)d_out;

    const int N = in_sizes[0] / F_ATOM;   // 50000
    const int E = in_sizes[1] / 2;        // 600000
    const int P = in_sizes[3] / F_PAIR;   // 200000
    const int* row = eidx;
    const int* col = eidx + E;

    // workspace carve-up
    char* ws = (char*)d_ws;
    size_t off = 0;
    float* deg      = (float*)(ws + off); off = align256(off + (size_t)N * sizeof(float));
    float* dinv     = (float*)(ws + off); off = align256(off + (size_t)N * sizeof(float));
    int* rowstart   = (int*)(ws + off);   off = align256(off + (size_t)(N + 1) * sizeof(int));
    int* cursor     = (int*)(ws + off);   off = align256(off + (size_t)N * sizeof(int));
    int* col_sorted = (int*)(ws + off);   off = align256(off + (size_t)(E + N) * sizeof(int));
    float* x        = (float*)(ws + off); off = align256(off + (size_t)N * HID * sizeof(float));
    float* h        = (float*)(ws + off); off = align256(off + (size_t)N * HID * sizeof(float));
    (void)ws_size;

    const int B = 256;
    // 1) degree + normalization
    init_deg_kernel<<<(N + B - 1) / B, B, 0, stream>>>(deg, N);
    count_deg_kernel<<<(E + B - 1) / B, B, 0, stream>>>(row, deg, E);
    dinv_kernel<<<(N + B - 1) / B, B, 0, stream>>>(deg, dinv, N);

    // 2) CSR build (self-loop first in each row)
    scan_kernel<<<1, 1024, 0, stream>>>(deg, rowstart, N);
    csr_selfloop_kernel<<<(N + B - 1) / B, B, 0, stream>>>(rowstart, cursor, col_sorted, N);
    csr_fill_kernel<<<(E + B - 1) / B, B, 0, stream>>>(row, col, rowstart, cursor, col_sorted, E);

    // 3) embed: x = atom @ embed_W + embed_b   (WMMA f32)
    gemm_xw_kernel<<<N / 16, B, 0, stream>>>(atom, emW, emB, x, F_ATOM);

    // 4) GCN layers: h = x @ W; x = relu(BN(aggregate(h) + b))
    for (int L = 0; L < N_LAYERS; ++L) {
        gemm_xw_kernel<<<N / 16, B, 0, stream>>>(x, convW + (size_t)L * HID * HID,
                                                 nullptr, h, HID);
        aggregate_bn_relu_kernel<<<(N * 32 + B - 1) / B, B, 0, stream>>>(
            h, rowstart, col_sorted, dinv,
            convB + (size_t)L * HID, gamma + (size_t)L * HID, beta + (size_t)L * HID,
            mean + (size_t)L * HID, var + (size_t)L * HID, x, N);
    }

    // 5) fused pair MLP
    pair_mlp_kernel<<<P / 16, B, 0, stream>>>(x, pairs, pfeat, W1, b1, W2, b2, W3, b3, out);
    (void)n_in; (void)out_size;
}